// IDCT2_80599356277184
// MI455X (gfx1250) — compile-verified
//
#include <hip/hip_runtime.h>

// 2D IDCT (inverse of unnormalized DCT-II) as two f32 WMMA GEMM passes:
//   B[m,k] = (1/n) * w_k * cos(pi*k*(2m+1)/(2n)),  w_0 = 1/2
//   op(P)  = B * P^T   (row-major 4096x4096)
//   out    = op(op(x)) = B * x * B^T
//
// Each wave computes a 64x64 output tile as 4x4 grid of 16x16 WMMA tiles,
// K-loop in steps of 4 using V_WMMA_F32_16X16X4_F32 (full fp32 precision).

#define NDIM  4096
#define LOG2N 12

typedef float v2f __attribute__((ext_vector_type(2)));
typedef float v8f __attribute__((ext_vector_type(8)));

// ---------------------------------------------------------------------------
// Basis precompute: B[m*N + k] = (1/n)*w_k*cospi( ((k*(2m+1)) mod 4n) / (2n) )
// 4n = 16384 = 2^14 -> exact integer reduction; t/8192 is exact in f32.
// ---------------------------------------------------------------------------
__global__ __launch_bounds__(256) void idct_basis_kernel(float* __restrict__ B) {
    int idx = blockIdx.x * 256 + threadIdx.x;
    int m = idx >> LOG2N;
    int k = idx & (NDIM - 1);
    int t = (k * (2 * m + 1)) & (4 * NDIM - 1);
    float w = (k == 0) ? (0.5f / NDIM) : (1.0f / NDIM);
    B[idx] = w * cospif((float)t * (1.0f / (2.0f * NDIM)));
}

__device__ __forceinline__ float basis_val(int m, int k) {
    int t = (k * (2 * m + 1)) & (4 * NDIM - 1);
    float w = (k == 0) ? (0.5f / NDIM) : (1.0f / NDIM);
    return w * cospif((float)t * (1.0f / (2.0f * NDIM)));
}

// ---------------------------------------------------------------------------
// Z = B * P^T : Z[m][i] = sum_k B[m][k] * P[i][k]
// Block: 256 threads = 8 waves (4 along m, 2 along i); block tile 256(m)x128(i)
// Wave tile: 64x64 = 4x4 WMMA 16x16 tiles.
// FLY=true computes basis fragments on the fly (small-ws fallback).
// ---------------------------------------------------------------------------
template <bool FLY>
__global__ __launch_bounds__(256) void idct_gemm_kernel(const float* __restrict__ Bmat,
                                                        const float* __restrict__ P,
                                                        float* __restrict__ Z) {
    const int lane = threadIdx.x & 31;
    const int wv   = threadIdx.x >> 5;
    const int wm   = wv & 3;        // 4 waves along m
    const int wi   = wv >> 2;       // 2 waves along i
    const int l16  = lane & 15;
    const int hi   = lane >> 4;     // half-wave selects K pair {0,1} vs {2,3}

    const int m_base = blockIdx.y * 256 + wm * 64;
    const int i_base = blockIdx.x * 128 + wi * 64;

    // Per-lane row pointers: both A and B fragments are float2 loads at
    // ptr[(row_base + l16)*N + k0 + 2*hi]  (8-byte aligned: k0 % 4 == 0).
    const float* aptr[4];
    const float* bptr[4];
#pragma unroll
    for (int t = 0; t < 4; ++t)
        aptr[t] = FLY ? nullptr
                      : (Bmat + (size_t)(m_base + t * 16 + l16) * NDIM + 2 * hi);
#pragma unroll
    for (int u = 0; u < 4; ++u)
        bptr[u] = P + (size_t)(i_base + u * 16 + l16) * NDIM + 2 * hi;

    v8f acc[4][4];
#pragma unroll
    for (int t = 0; t < 4; ++t)
#pragma unroll
        for (int u = 0; u < 4; ++u) {
            v8f z{};
            acc[t][u] = z;
        }

#pragma unroll 2
    for (int k0 = 0; k0 < NDIM; k0 += 4) {
        v2f a[4], b[4];
#pragma unroll
        for (int t = 0; t < 4; ++t) {
            if (FLY) {
                const int m  = m_base + t * 16 + l16;
                const int ka = k0 + 2 * hi;
                a[t][0] = basis_val(m, ka);
                a[t][1] = basis_val(m, ka + 1);
            } else {
                a[t] = *(const v2f*)(aptr[t] + k0);
            }
        }
#pragma unroll
        for (int u = 0; u < 4; ++u)
            b[u] = *(const v2f*)(bptr[u] + k0);

#pragma unroll
        for (int t = 0; t < 4; ++t)
#pragma unroll
            for (int u = 0; u < 4; ++u)
                acc[t][u] = __builtin_amdgcn_wmma_f32_16x16x4_f32(
                    /*neg_a=*/false, a[t], /*neg_b=*/false, b[u],
                    /*c_mod=*/(short)0, acc[t][u],
                    /*reuse_a=*/false, /*reuse_b=*/false);
    }

    // D layout: VGPR r -> row (r + 8*hi), col l16 within each 16x16 tile.
#pragma unroll
    for (int t = 0; t < 4; ++t) {
#pragma unroll
        for (int u = 0; u < 4; ++u) {
            const int    col  = i_base + u * 16 + l16;
            const size_t rowb = (size_t)(m_base + t * 16 + hi * 8) * NDIM + col;
#pragma unroll
            for (int r = 0; r < 8; ++r)
                Z[rowb + (size_t)r * NDIM] = acc[t][u][r];
        }
    }
}

// ---------------------------------------------------------------------------
extern "C" void kernel_launch(void* const* d_in, const int* in_sizes, int n_in,
                              void* d_out, int out_size, void* d_ws, size_t ws_size,
                              hipStream_t stream) {
    const float* x   = (const float*)d_in[0];   // 4096x4096 f32
    // d_in[1] (expkM) and d_in[2] (expkN) unused: basis rebuilt exactly on device.
    float*       out = (float*)d_out;
    float*       ws  = (float*)d_ws;

    const size_t elems = (size_t)NDIM * NDIM;
    dim3 block(256);
    dim3 grid(NDIM / 128, NDIM / 256);          // 32 x 16 = 512 workgroups

    if (ws_size >= 2 * elems * sizeof(float)) {
        // ws: [basis 64MB][intermediate 64MB]
        float* Bpre = ws;
        float* Y1   = ws + elems;
        idct_basis_kernel<<<dim3((unsigned)(elems / 256)), block, 0, stream>>>(Bpre);
        idct_gemm_kernel<false><<<grid, block, 0, stream>>>(Bpre, x,  Y1);   // Y1 = B*x^T
        idct_gemm_kernel<false><<<grid, block, 0, stream>>>(Bpre, Y1, out);  // out = B*Y1^T = B*x*B^T
    } else {
        // Small-scratch fallback: basis fragments computed on the fly.
        float* Y1 = ws;                          // needs 64MB
        idct_gemm_kernel<true><<<grid, block, 0, stream>>>(nullptr, x,  Y1);
        idct_gemm_kernel<true><<<grid, block, 0, stream>>>(nullptr, Y1, out);
    }
}